// CustomSoftmaxExperts_47571057771179
// MI455X (gfx1250) — compile-verified
//
#include <hip/hip_runtime.h>
#include <stdint.h>

// ---------------------------------------------------------------------------
// Softmax(64) + top-5 threshold for (32, 8192, 64) fp32.
// mask = (s >= kth5) & (s >= 0.2) == (s >= 0.2): at most five 64-way softmax
// values can be >= 0.2 (six would sum past 1), so any s >= 0.2 is in the top 5.
// => kernel = row softmax + threshold. Pure HBM stream: 128 MB @ 23.3 TB/s.
//
// Layout: THREAD-per-row (no cross-lane shuffles at all).
//   * group = 32 rows (one per lane) = 8 KB, async-staged to LDS:
//       16 x global_load_async_to_lds_b128 per wave per group (ASYNCcnt)
//   * LDS rows padded to 68 floats: lane t reads its row with ds_load_b128,
//     banks (4t+4k) mod 64 -> benign 2-way conflict = full LDS rate
//   * per-wave private double buffer -> no __syncthreads; per-wave pipeline
//     controlled by s_wait_asynccnt / s_wait_dscnt
//   * reductions are in-register trees; exp via v_exp_f32 (TRANS co-issues)
// ---------------------------------------------------------------------------

#define THRESH           0.2f
#define COLS             64
#define BLOCK_THREADS    256
#define WAVES_PER_BLOCK  (BLOCK_THREADS / 32)          // 8
#define ROWS_PER_GROUP   32                            // one row per lane
#define GROUP_IN_BYTES   (ROWS_PER_GROUP * COLS * 4)   // 8192
#define CHUNKS           16                            // 16B chunks per lane/group
#define LDS_ROW_FLOATS   68                            // 64 + 4 pad
#define LDS_ROW_BYTES    (LDS_ROW_FLOATS * 4)          // 272
#define LDS_GROUP_FLOATS (ROWS_PER_GROUP * LDS_ROW_FLOATS) // 2176
#define LOG2E            1.4426950408889634f

// Issue one group's async loads: 16 chunks/lane-iteration, global side linear,
// LDS side scattered into the padded layout (chunk never crosses a row).
__device__ __forceinline__ void issue_group_loads(const float* __restrict__ in,
                                                  long g, uint32_t ldsLaneAddr,
                                                  int lane) {
    const char* gp = (const char*)in + (size_t)g * GROUP_IN_BYTES + (size_t)lane * 16;
#pragma unroll
    for (int j = 0; j < CHUNKS; ++j) {
        uint64_t ga = (uint64_t)(uintptr_t)gp + (uint32_t)(j * 512);       // +2 rows global
        uint32_t la = ldsLaneAddr + (uint32_t)(j * 2 * LDS_ROW_BYTES);     // +2 rows padded
        asm volatile("global_load_async_to_lds_b128 %0, %1, off"
                     :: "v"(la), "v"(ga) : "memory");
    }
}

__global__ __launch_bounds__(BLOCK_THREADS)
void moe_softmax_thresh_kernel(const float* __restrict__ in,
                               float* __restrict__ out,
                               long nGroups, int groupsPerWave) {
    // per-wave private double buffers: 8 * 2 * 2176 * 4 = 136 KB
    __shared__ float sh[WAVES_PER_BLOCK][2][LDS_GROUP_FLOATS];

    const int wave = threadIdx.x >> 5;
    const int lane = threadIdx.x & 31;
    const long W   = (long)blockIdx.x * WAVES_PER_BLOCK + wave;

    long g0 = W * groupsPerWave;
    long g1 = g0 + groupsPerWave;
    if (g1 > nGroups) g1 = nGroups;
    if (g0 >= g1) return;                 // whole-wave uniform exit

    // this lane's chunk-0 LDS offset in the padded layout:
    // chunk c = j*32+lane -> row 2j+(lane>>4), 16B-slot (lane&15)
    const uint32_t ldsLane  = (uint32_t)((lane >> 4) * LDS_ROW_BYTES + (lane & 15) * 16);
    const uint32_t ldsBufA  = (uint32_t)(uintptr_t)&sh[wave][0][0] + ldsLane;
    const uint32_t ldsBufB  = (uint32_t)(uintptr_t)&sh[wave][1][0] + ldsLane;

    // prologue: prefetch up to two groups (independent per-wave pipeline)
    issue_group_loads(in, g0, ldsBufA, lane);
    if (g0 + 1 < g1) issue_group_loads(in, g0 + 1, ldsBufB, lane);

    for (long g = g0; g < g1; ++g) {
        const int buf = (int)(g - g0) & 1;

        // wait for THIS group's 16 chunk-loads (async loads complete in order;
        // only the next group's 16 may still be outstanding)
        if (g + 1 < g1) asm volatile("s_wait_asynccnt 0x10" ::: "memory");
        else            asm volatile("s_wait_asynccnt 0x0"  ::: "memory");

        // pull this lane's whole row (64 floats) into registers
        const char* rowBytes = (const char*)&sh[wave][buf][0]
                             + (size_t)lane * LDS_ROW_BYTES;
        float4 xv[CHUNKS];
#pragma unroll
        for (int k = 0; k < CHUNKS; ++k)
            xv[k] = ((const float4*)rowBytes)[k];

        // LDS reads must land before this buffer is overwritten by g+2 prefetch
        asm volatile("s_wait_dscnt 0x0" ::: "memory");
        if (g + 2 < g1)
            issue_group_loads(in, g + 2, (buf == 0) ? ldsBufA : ldsBufB, lane);

        // ---- row max: 4-wide accumulator tree, no cross-lane traffic ----
        float4 m4 = xv[0];
#pragma unroll
        for (int k = 1; k < CHUNKS; ++k) {
            m4.x = fmaxf(m4.x, xv[k].x); m4.y = fmaxf(m4.y, xv[k].y);
            m4.z = fmaxf(m4.z, xv[k].z); m4.w = fmaxf(m4.w, xv[k].w);
        }
        const float m   = fmaxf(fmaxf(m4.x, m4.y), fmaxf(m4.z, m4.w));
        const float nmk = -m * LOG2E;     // exp(x-m) == exp2(x*log2e - m*log2e)

        // ---- exp + sum (1 fma + 1 v_exp + 1 add per element) ----
        float4 s4 = make_float4(0.f, 0.f, 0.f, 0.f);
#pragma unroll
        for (int k = 0; k < CHUNKS; ++k) {
            xv[k].x = __builtin_amdgcn_exp2f(fmaf(xv[k].x, LOG2E, nmk));
            xv[k].y = __builtin_amdgcn_exp2f(fmaf(xv[k].y, LOG2E, nmk));
            xv[k].z = __builtin_amdgcn_exp2f(fmaf(xv[k].z, LOG2E, nmk));
            xv[k].w = __builtin_amdgcn_exp2f(fmaf(xv[k].w, LOG2E, nmk));
            s4.x += xv[k].x; s4.y += xv[k].y; s4.z += xv[k].z; s4.w += xv[k].w;
        }
        const float s   = (s4.x + s4.y) + (s4.z + s4.w);
        const float inv = __builtin_amdgcn_rcpf(s);

        // ---- normalize + threshold + store (b128 per thread) ----
        float4* orow = (float4*)(out + ((size_t)g * ROWS_PER_GROUP + lane) * COLS);
#pragma unroll
        for (int k = 0; k < CHUNKS; ++k) {
            float4 p;
            p.x = xv[k].x * inv; p.x = (p.x >= THRESH) ? p.x : 0.0f;
            p.y = xv[k].y * inv; p.y = (p.y >= THRESH) ? p.y : 0.0f;
            p.z = xv[k].z * inv; p.z = (p.z >= THRESH) ? p.z : 0.0f;
            p.w = xv[k].w * inv; p.w = (p.w >= THRESH) ? p.w : 0.0f;
            orow[k] = p;
        }
    }
}

// ---------------------------------------------------------------------------
// Tail (rows % 32 != 0; unused for this shape): wave-per-row with shuffles.
// ---------------------------------------------------------------------------
__device__ __forceinline__ float wave_reduce_max(float v) {
#pragma unroll
    for (int off = 16; off > 0; off >>= 1)
        v = fmaxf(v, __shfl_xor(v, off, 32));
    return v;
}
__device__ __forceinline__ float wave_reduce_sum(float v) {
#pragma unroll
    for (int off = 16; off > 0; off >>= 1)
        v += __shfl_xor(v, off, 32);
    return v;
}

__global__ __launch_bounds__(BLOCK_THREADS)
void moe_softmax_thresh_tail_kernel(const float* __restrict__ in,
                                    float* __restrict__ out,
                                    long startRow, long endRow) {
    const long row = startRow + (long)blockIdx.x * WAVES_PER_BLOCK + (threadIdx.x >> 5);
    if (row >= endRow) return;
    const int lane = threadIdx.x & 31;

    const float2* ip = (const float2*)(in + (size_t)row * COLS);
    float2 v = ip[lane];

    float m  = wave_reduce_max(fmaxf(v.x, v.y));
    float nmk = -m * LOG2E;
    float e0 = __builtin_amdgcn_exp2f(fmaf(v.x, LOG2E, nmk));
    float e1 = __builtin_amdgcn_exp2f(fmaf(v.y, LOG2E, nmk));
    float s  = wave_reduce_sum(e0 + e1);
    float inv = __builtin_amdgcn_rcpf(s);
    float p0 = e0 * inv; p0 = (p0 >= THRESH) ? p0 : 0.0f;
    float p1 = e1 * inv; p1 = (p1 >= THRESH) ? p1 : 0.0f;

    float2* op = (float2*)(out + (size_t)row * COLS);
    op[lane] = make_float2(p0, p1);
}

extern "C" void kernel_launch(void* const* d_in, const int* in_sizes, int n_in,
                              void* d_out, int out_size, void* d_ws, size_t ws_size,
                              hipStream_t stream) {
    const float* in  = (const float*)d_in[0];
    float*       out = (float*)d_out;

    const long R       = (long)in_sizes[0] / COLS;     // 262144 rows
    const long nGroups = R / ROWS_PER_GROUP;           // 8192 groups

    if (nGroups > 0) {
        const long maxBlocks  = 512;                   // 4096 waves in flight
        const long totalWaves = maxBlocks * WAVES_PER_BLOCK;
        const int  gpw        = (int)((nGroups + totalWaves - 1) / totalWaves);
        const long wavesNeeded = (nGroups + gpw - 1) / gpw;
        const long blocks      = (wavesNeeded + WAVES_PER_BLOCK - 1) / WAVES_PER_BLOCK;
        moe_softmax_thresh_kernel<<<(int)blocks, BLOCK_THREADS, 0, stream>>>(
            in, out, nGroups, gpw);
    }

    const long startRow = nGroups * ROWS_PER_GROUP;
    if (startRow < R) {
        const long tailRows = R - startRow;
        const long blocks   = (tailRows + WAVES_PER_BLOCK - 1) / WAVES_PER_BLOCK;
        moe_softmax_thresh_tail_kernel<<<(int)blocks, BLOCK_THREADS, 0, stream>>>(
            in, out, startRow, R);
    }
}